// GATEmbedding_86002425135777
// MI455X (gfx1250) — compile-verified
//
#include <hip/hip_runtime.h>
#include <math.h>

#define NNODES 50000
#define NEDGES 400000
#define EP     (NEDGES + NNODES)   // edges + self loops
#define HH     4
#define CC     64
#define DD1    256                 // H*C
#define IN0    36
#define OUTD   128

typedef float v2f __attribute__((ext_vector_type(2)));
typedef float v8f __attribute__((ext_vector_type(8)));

// ---------------------------------------------------------------- utilities
__device__ __forceinline__ float waveSum(float v) {
  v += __shfl_xor(v, 16, 32);
  v += __shfl_xor(v, 8, 32);
  v += __shfl_xor(v, 4, 32);
  v += __shfl_xor(v, 2, 32);
  v += __shfl_xor(v, 1, 32);
  return v;
}

__device__ __forceinline__ void atomicMaxFloat(float* addr, float v) {
  // order-preserving bit trick: signed max for >=0, unsigned min for <0
  if (v >= 0.f) atomicMax((int*)addr, __float_as_int(v));
  else          atomicMin((unsigned int*)addr, (unsigned int)__float_as_int(v));
}

__global__ void fill_kernel(float* __restrict__ p, float v, int n) {
  int i = blockIdx.x * blockDim.x + threadIdx.x;
  if (i < n) p[i] = v;
}

// ---------------------------------------------------------------- embedding
// h0[i] = concat(x[i,0:4], emb_table[joint_types[i]])  -> [N, 36]
__global__ void embed_kernel(const float* __restrict__ x, const int* __restrict__ jt,
                             const float* __restrict__ emb, float* __restrict__ h0) {
  int idx = blockIdx.x * blockDim.x + threadIdx.x;
  if (idx >= NNODES * IN0) return;
  int i = idx / IN0, j = idx - i * IN0;
  h0[idx] = (j < 4) ? x[i * 4 + j] : emb[jt[i] * 32 + (j - 4)];
}

// ---------------------------------------------------------------- WMMA GEMM
// C[M,Nout] = A[M,K] @ W[K,Nout] + bias[Nout]
// fp32 WMMA 16x16x4, one wave per 16x16 tile, 4 waves/block (64 output cols).
// M % 16 == 0, K % 4 == 0, Nout % 64 == 0 (true for all uses here).
__global__ void gemm_wmma_kernel(const float* __restrict__ A, const float* __restrict__ W,
                                 const float* __restrict__ bias, float* __restrict__ Cout,
                                 int M, int K, int Nout) {
  const int lane = threadIdx.x & 31;
  const int wave = threadIdx.x >> 5;
  const int m0 = blockIdx.x * 16;
  const int n0 = (blockIdx.y * 4 + wave) * 16;
  const int half = lane >> 4;        // 0: lanes 0-15, 1: lanes 16-31
  const int l15  = lane & 15;

  const float* arow = A + (size_t)(m0 + l15) * K;
  const float* wcol = W + n0 + l15;

  v8f acc = {};
  for (int kk = 0; kk < K; kk += 4) {
    v2f a, b;
    int ak = kk + half * 2;          // A 16x4 frag: lanes0-15 K={0,1}, lanes16-31 K={2,3}
    a.x = arow[ak];
    a.y = arow[ak + 1];
    b.x = wcol[(size_t)ak * Nout];   // B 4x16 frag mirrors the A K-split
    b.y = wcol[(size_t)(ak + 1) * Nout];
    acc = __builtin_amdgcn_wmma_f32_16x16x4_f32(false, a, false, b, (short)0, acc,
                                                false, false);
  }

  const float bv = bias[n0 + l15];
  const int col = n0 + l15;
  const int rbase = m0 + half * 8;   // C 16x16: VGPR i -> rows i (lo half) / i+8 (hi half)
#pragma unroll
  for (int i = 0; i < 8; i++)
    Cout[(size_t)(rbase + i) * Nout + col] = acc[i] + bv;
}

// ---------------------------------------------------------------- edge score
// One wave per edge. lane -> head = lane>>3, 8 channels per lane.
// score[e,h] = sum_c att[h,c] * leaky_relu(xl[src,h,c] + xr[dst,h,c])
__global__ void edge_score_kernel(const int* __restrict__ ei,
                                  const float* __restrict__ xl, const float* __restrict__ xr,
                                  const float* __restrict__ att,
                                  float* __restrict__ score, float* __restrict__ smax) {
  int e = (blockIdx.x * blockDim.x + threadIdx.x) >> 5;
  int lane = threadIdx.x & 31;
  if (e >= EP) return;
  int src = (e < NEDGES) ? ei[e] : (e - NEDGES);
  int dst = (e < NEDGES) ? ei[NEDGES + e] : (e - NEDGES);
  int h = lane >> 3;
  int cbase = h * CC + (lane & 7) * 8;
  const float* pl = xl + (size_t)src * DD1 + cbase;
  const float* pr = xr + (size_t)dst * DD1 + cbase;
  const float* pa = att + cbase;
  float s = 0.f;
#pragma unroll
  for (int k = 0; k < 8; k++) {
    float v = pl[k] + pr[k];
    v = (v > 0.f) ? v : 0.2f * v;    // leaky_relu(0.2)
    s += pa[k] * v;
  }
  s += __shfl_xor(s, 1, 32);
  s += __shfl_xor(s, 2, 32);
  s += __shfl_xor(s, 4, 32);
  if ((lane & 7) == 0) {
    score[(size_t)e * HH + h] = s;
    atomicMaxFloat(smax + (size_t)dst * HH + h, s);
  }
}

// exp(score - smax[dst]) in place + denominator accumulation
__global__ void edge_exp_kernel(const int* __restrict__ ei, const float* __restrict__ smax,
                                float* __restrict__ score, float* __restrict__ denom) {
  int idx = blockIdx.x * blockDim.x + threadIdx.x;
  if (idx >= EP * HH) return;
  int e = idx >> 2, h = idx & 3;
  int dst = (e < NEDGES) ? ei[NEDGES + e] : (e - NEDGES);
  float ex = __expf(score[idx] - smax[(size_t)dst * HH + h]);
  score[idx] = ex;
  atomicAdd(denom + (size_t)dst * HH + h, ex);
}

// out[dst] += alpha * xl[src]   (block = one edge, 256 coalesced f32 atomics)
__global__ void edge_scatter_kernel(const int* __restrict__ ei, const float* __restrict__ xl,
                                    const float* __restrict__ score, const float* __restrict__ denom,
                                    float* __restrict__ outb) {
  int e = blockIdx.x;
  int j = threadIdx.x;               // 0..255
  int src = (e < NEDGES) ? ei[e] : (e - NEDGES);
  int dst = (e < NEDGES) ? ei[NEDGES + e] : (e - NEDGES);
  int h = j >> 6;
  float alpha = score[(size_t)e * HH + h] / denom[(size_t)dst * HH + h];
  atomicAdd(outb + (size_t)dst * DD1 + j, alpha * xl[(size_t)src * DD1 + j]);
}

// ---------------------------------------------------------------- node finalize
// layer0: in-place  h = elu(layernorm(out + bo, g, beta))  over 256 dims, wave/node
__global__ void finalize0_kernel(float* __restrict__ outb, const float* __restrict__ bo,
                                 const float* __restrict__ g, const float* __restrict__ beta) {
  int node = (blockIdx.x * blockDim.x + threadIdx.x) >> 5;
  int lane = threadIdx.x & 31;
  if (node >= NNODES) return;
  float v[8], s = 0.f, sq = 0.f;
#pragma unroll
  for (int k = 0; k < 8; k++) {
    int j = lane + 32 * k;
    v[k] = outb[(size_t)node * DD1 + j] + bo[j];
    s += v[k]; sq += v[k] * v[k];
  }
  s = waveSum(s); sq = waveSum(sq);
  float mean = s * (1.f / 256.f);
  float var = sq * (1.f / 256.f) - mean * mean;
  float inv = rsqrtf(var + 1e-5f);
#pragma unroll
  for (int k = 0; k < 8; k++) {
    int j = lane + 32 * k;
    float y = (v[k] - mean) * inv * g[j] + beta[j];
    outb[(size_t)node * DD1 + j] = (y > 0.f) ? y : expm1f(y);   // elu
  }
}

// layer1: t1 = elu(layernorm(mean_heads(out) + bo, g, beta))  over 64 dims
__global__ void finalize1_kernel(const float* __restrict__ outb, const float* __restrict__ bo,
                                 const float* __restrict__ g, const float* __restrict__ beta,
                                 float* __restrict__ t1) {
  int node = (blockIdx.x * blockDim.x + threadIdx.x) >> 5;
  int lane = threadIdx.x & 31;
  if (node >= NNODES) return;
  float v[2], s = 0.f, sq = 0.f;
#pragma unroll
  for (int k = 0; k < 2; k++) {
    int c = lane + 32 * k;
    const float* p = outb + (size_t)node * DD1 + c;
    float m = 0.25f * (p[0] + p[64] + p[128] + p[192]) + bo[c];
    v[k] = m; s += m; sq += m * m;
  }
  s = waveSum(s); sq = waveSum(sq);
  float mean = s * (1.f / 64.f);
  float var = sq * (1.f / 64.f) - mean * mean;
  float inv = rsqrtf(var + 1e-5f);
#pragma unroll
  for (int k = 0; k < 2; k++) {
    int c = lane + 32 * k;
    float y = (v[k] - mean) * inv * g[c] + beta[c];
    t1[(size_t)node * CC + c] = (y > 0.f) ? y : expm1f(y);
  }
}

// final: out = L2normalize(layernorm(proj, gf, betaf))  over 128 dims
__global__ void finalf_kernel(const float* __restrict__ proj, const float* __restrict__ g,
                              const float* __restrict__ beta, float* __restrict__ out) {
  int node = (blockIdx.x * blockDim.x + threadIdx.x) >> 5;
  int lane = threadIdx.x & 31;
  if (node >= NNODES) return;
  float v[4], s = 0.f, sq = 0.f;
#pragma unroll
  for (int k = 0; k < 4; k++) {
    int j = lane + 32 * k;
    v[k] = proj[(size_t)node * OUTD + j];
    s += v[k]; sq += v[k] * v[k];
  }
  s = waveSum(s); sq = waveSum(sq);
  float mean = s * (1.f / 128.f);
  float var = sq * (1.f / 128.f) - mean * mean;
  float inv = rsqrtf(var + 1e-5f);
  float y[4], nsq = 0.f;
#pragma unroll
  for (int k = 0; k < 4; k++) {
    int j = lane + 32 * k;
    y[k] = (v[k] - mean) * inv * g[j] + beta[j];
    nsq += y[k] * y[k];
  }
  nsq = waveSum(nsq);
  float scale = 1.f / fmaxf(sqrtf(nsq), 1e-12f);
#pragma unroll
  for (int k = 0; k < 4; k++)
    out[(size_t)node * OUTD + lane + 32 * k] = y[k] * scale;
}

// ---------------------------------------------------------------- launcher
extern "C" void kernel_launch(void* const* d_in, const int* in_sizes, int n_in,
                              void* d_out, int out_size, void* d_ws, size_t ws_size,
                              hipStream_t stream) {
  const float* x     = (const float*)d_in[0];
  const int*   jt    = (const int*)  d_in[1];
  const int*   ei    = (const int*)  d_in[2];   // [2,E]: rows = src, dst
  const float* emb   = (const float*)d_in[3];
  const float* Wl0   = (const float*)d_in[4];
  const float* bl0   = (const float*)d_in[5];
  const float* Wr0   = (const float*)d_in[6];
  const float* br0   = (const float*)d_in[7];
  const float* att0  = (const float*)d_in[8];
  const float* bo0   = (const float*)d_in[9];
  const float* g0    = (const float*)d_in[10];
  const float* beta0 = (const float*)d_in[11];
  const float* Wl1   = (const float*)d_in[12];
  const float* bl1   = (const float*)d_in[13];
  const float* Wr1   = (const float*)d_in[14];
  const float* br1   = (const float*)d_in[15];
  const float* att1  = (const float*)d_in[16];
  const float* bo1   = (const float*)d_in[17];
  const float* g1    = (const float*)d_in[18];
  const float* beta1 = (const float*)d_in[19];
  const float* Wp    = (const float*)d_in[20];
  const float* bp    = (const float*)d_in[21];
  const float* gf    = (const float*)d_in[22];
  const float* betaf = (const float*)d_in[23];

  float* ws = (float*)d_ws;
  float* h0     = ws;                               // N*36
  float* xl     = h0     + (size_t)NNODES * IN0;    // N*256
  float* xr     = xl     + (size_t)NNODES * DD1;    // N*256
  float* outb   = xr     + (size_t)NNODES * DD1;    // N*256 (accumulator, reused)
  float* scoreb = outb   + (size_t)NNODES * DD1;    // EP*4
  float* smax   = scoreb + (size_t)EP * HH;         // N*4
  float* denom  = smax   + (size_t)NNODES * HH;     // N*4
  float* t1   = xl;                                 // reuse after layer1 scatter
  float* proj = xr;                                 // reuse after layer1 scatter

  const float NEGBIG = -3.402823466e38f;

  // ---- embedding
  embed_kernel<<<(NNODES * IN0 + 255) / 256, 256, 0, stream>>>(x, jt, emb, h0);

  // ---- layer 0: node transforms (fp32 WMMA)
  gemm_wmma_kernel<<<dim3(NNODES / 16, DD1 / 64), 128, 0, stream>>>(h0, Wl0, bl0, xl, NNODES, IN0, DD1);
  gemm_wmma_kernel<<<dim3(NNODES / 16, DD1 / 64), 128, 0, stream>>>(h0, Wr0, br0, xr, NNODES, IN0, DD1);

  // ---- layer 0: edge softmax + aggregation
  fill_kernel<<<(NNODES * HH + 255) / 256, 256, 0, stream>>>(smax, NEGBIG, NNODES * HH);
  fill_kernel<<<(NNODES * HH + 255) / 256, 256, 0, stream>>>(denom, 0.f, NNODES * HH);
  fill_kernel<<<(NNODES * DD1 + 255) / 256, 256, 0, stream>>>(outb, 0.f, NNODES * DD1);
  edge_score_kernel<<<(EP * 32 + 255) / 256, 256, 0, stream>>>(ei, xl, xr, att0, scoreb, smax);
  edge_exp_kernel<<<(EP * HH + 255) / 256, 256, 0, stream>>>(ei, smax, scoreb, denom);
  edge_scatter_kernel<<<EP, 256, 0, stream>>>(ei, xl, scoreb, denom, outb);
  finalize0_kernel<<<(NNODES * 32 + 255) / 256, 256, 0, stream>>>(outb, bo0, g0, beta0);

  // ---- layer 1: node transforms on the in-place activations
  gemm_wmma_kernel<<<dim3(NNODES / 16, DD1 / 64), 128, 0, stream>>>(outb, Wl1, bl1, xl, NNODES, DD1, DD1);
  gemm_wmma_kernel<<<dim3(NNODES / 16, DD1 / 64), 128, 0, stream>>>(outb, Wr1, br1, xr, NNODES, DD1, DD1);

  // ---- layer 1: edge softmax + aggregation
  fill_kernel<<<(NNODES * HH + 255) / 256, 256, 0, stream>>>(smax, NEGBIG, NNODES * HH);
  fill_kernel<<<(NNODES * HH + 255) / 256, 256, 0, stream>>>(denom, 0.f, NNODES * HH);
  fill_kernel<<<(NNODES * DD1 + 255) / 256, 256, 0, stream>>>(outb, 0.f, NNODES * DD1);
  edge_score_kernel<<<(EP * 32 + 255) / 256, 256, 0, stream>>>(ei, xl, xr, att1, scoreb, smax);
  edge_exp_kernel<<<(EP * HH + 255) / 256, 256, 0, stream>>>(ei, smax, scoreb, denom);
  edge_scatter_kernel<<<EP, 256, 0, stream>>>(ei, xl, scoreb, denom, outb);
  finalize1_kernel<<<(NNODES * 32 + 255) / 256, 256, 0, stream>>>(outb, bo1, g1, beta1, t1);

  // ---- projection + final LN + L2 normalize
  gemm_wmma_kernel<<<dim3(NNODES / 16, OUTD / 64), 128, 0, stream>>>(t1, Wp, bp, proj, NNODES, CC, OUTD);
  finalf_kernel<<<(NNODES * 32 + 255) / 256, 256, 0, stream>>>(proj, gf, betaf, (float*)d_out);
}